// Attention_68624987455753
// MI455X (gfx1250) — compile-verified
//
#include <hip/hip_runtime.h>

typedef __attribute__((ext_vector_type(2))) float v2f;
typedef __attribute__((ext_vector_type(4))) float v4f;
typedef __attribute__((ext_vector_type(8))) float v8f;

#define SS 512
#define DD 64
#define HH 8

// One workgroup = one 16-row q-tile x 2 heads. 8 waves:
//   waves 0-3 -> head pair member 0, waves 4-7 -> head pair member 1.
// relem is loaded ONCE per (q,k,d) and applied to both heads (phases B/E),
// halving L2 traffic on the 64MB relem tensor vs one-head-per-block.
__global__ __launch_bounds__(256) void attn_relem_kernel(
    const float* __restrict__ Q, const float* __restrict__ K,
    const float* __restrict__ V, const float* __restrict__ R,
    const int* __restrict__ M, float* __restrict__ out_ans,
    float* __restrict__ out_p)
{
    __shared__ float sc[2][16][SS];   // scores -> probabilities (64 KB)
    __shared__ float ab[2][16][DD];   // WMMA P@V partial tiles  (8 KB)

    const int qt   = blockIdx.x;        // q-tile: 0..31
    const int hp   = blockIdx.y;        // head pair: 0..3
    const int q0   = qt * 16;
    const int tid  = threadIdx.x;
    const int w    = tid >> 5;          // wave id 0..7
    const int hw   = w >> 2;            // which head of the pair this wave serves
    const int w2   = w & 3;             // wave id within head group
    const int lane = tid & 31;
    const int half = lane >> 4;         // 0: lanes 0-15, 1: lanes 16-31
    const int l15  = lane & 15;
    const int kofs = half * 2;          // K-offset inside 4-wide step

    const int h0 = hp * 2;              // heads h0, h0+1
    const int hme = h0 + hw;            // this wave's head for WMMA phases

    const float* Qh = Q + (size_t)hme * SS * DD;
    const float* Kh = K + (size_t)hme * SS * DD;
    const float* Vh = V + (size_t)hme * SS * DD;

    // ---- preload A fragments (Q tile, 16x64): qa[s] covers K = 4s..4s+3
    // ISA 32-bit A 16x4 layout: lane<16 -> (K=0,K=1); lane>=16 -> (K=2,K=3)
    v2f qa[16];
    {
        const float* qrow = Qh + (size_t)(q0 + l15) * DD;
        #pragma unroll
        for (int s = 0; s < 16; ++s) {
            qa[s].x = qrow[4 * s + kofs];
            qa[s].y = qrow[4 * s + kofs + 1];
        }
    }

    // ---- Phase A: scores = Q @ K^T via v_wmma_f32_16x16x4_f32
    for (int kt = w2; kt < SS / 16; kt += 4) {
        v8f acc = {};
        const float* krow = Kh + (size_t)(kt * 16 + l15) * DD;
        #pragma unroll
        for (int s = 0; s < 16; ++s) {
            v2f kb;
            kb.x = krow[4 * s + kofs];
            kb.y = krow[4 * s + kofs + 1];
            acc = __builtin_amdgcn_wmma_f32_16x16x4_f32(
                false, qa[s], false, kb, (short)0, acc, false, false);
        }
        // C/D layout: VGPR i -> row i (lanes 0-15) / row i+8 (lanes 16-31)
        #pragma unroll
        for (int i = 0; i < 8; ++i)
            sc[hw][i + half * 8][kt * 16 + l15] = acc[i];
    }
    __syncthreads();

    // ---- Phase B: add relem score term for BOTH heads (one relem read),
    //      then apply mask.
    for (int idx = tid; idx < 16 * SS; idx += 256) {
        const int qq = idx >> 9;
        const int kk = idx & (SS - 1);
        const float* rrow  = R + ((size_t)(q0 + qq) * SS + kk) * DD;
        const float* qrow0 = Q + ((size_t)h0 * SS + (q0 + qq)) * DD;
        const float* qrow1 = qrow0 + (size_t)SS * DD;
        float dot0 = 0.f, dot1 = 0.f;
        #pragma unroll 4
        for (int d4 = 0; d4 < DD; d4 += 4) {
            const v4f r  = *(const v4f*)(rrow + d4);
            const v4f a0 = *(const v4f*)(qrow0 + d4);
            const v4f a1 = *(const v4f*)(qrow1 + d4);
            dot0 += r.x * a0.x + r.y * a0.y + r.z * a0.z + r.w * a0.w;
            dot1 += r.x * a1.x + r.y * a1.y + r.z * a1.z + r.w * a1.w;
        }
        const bool dead = (M[(size_t)(q0 + qq) * SS + kk] == 0);
        float s0 = sc[0][qq][kk] + dot0;
        float s1 = sc[1][qq][kk] + dot1;
        sc[0][qq][kk] = dead ? -1e9f : s0;
        sc[1][qq][kk] = dead ? -1e9f : s1;
    }
    __syncthreads();

    // ---- Phase C: softmax over k per row (32 rows = 2 heads x 16);
    //      write p_attn to global.
    for (int r = w; r < 32; r += 8) {
        const int hh = r >> 4;
        const int rr = r & 15;
        float m = -INFINITY;
        for (int kk = lane; kk < SS; kk += 32) m = fmaxf(m, sc[hh][rr][kk]);
        #pragma unroll
        for (int off = 16; off > 0; off >>= 1)
            m = fmaxf(m, __shfl_xor(m, off, 32));
        float sum = 0.f;
        for (int kk = lane; kk < SS; kk += 32) {
            const float e = __expf(sc[hh][rr][kk] - m);
            sc[hh][rr][kk] = e;
            sum += e;
        }
        #pragma unroll
        for (int off = 16; off > 0; off >>= 1)
            sum += __shfl_xor(sum, off, 32);
        const float inv = 1.f / sum;
        float* pg = out_p + ((size_t)(h0 + hh) * SS + (q0 + rr)) * SS;
        for (int kk = lane; kk < SS; kk += 32) {
            const float p = sc[hh][rr][kk] * inv;
            sc[hh][rr][kk] = p;
            pg[kk] = p;
        }
    }
    __syncthreads();

    // ---- Phase D: ans_gemm = P @ V via WMMA; wave w owns
    //      head hw, columns n0..n0+15.
    {
        const int n0 = w2 * 16;
        v8f acc = {};
        for (int ks = 0; ks < SS / 4; ++ks) {
            const int kb = ks * 4;
            v2f a;                                     // A = P[16 x 4] chunk
            a.x = sc[hw][l15][kb + kofs];
            a.y = sc[hw][l15][kb + kofs + 1];
            v2f b;                                     // B = V[4 x 16] chunk
            b.x = Vh[(size_t)(kb + kofs) * DD + n0 + l15];
            b.y = Vh[(size_t)(kb + kofs + 1) * DD + n0 + l15];
            acc = __builtin_amdgcn_wmma_f32_16x16x4_f32(
                false, a, false, b, (short)0, acc, false, false);
        }
        #pragma unroll
        for (int i = 0; i < 8; ++i)
            ab[hw][i + half * 8][n0 + l15] = acc[i];
    }
    __syncthreads();

    // ---- Phase E: add relem output term for BOTH heads (one relem read),
    //      store ans.
    for (int idx = tid; idx < 16 * DD; idx += 256) {
        const int qq = idx >> 6;
        const int dd = idx & (DD - 1);
        const float* rcol = R + (size_t)(q0 + qq) * SS * DD + dd;
        float a00 = 0.f, a01 = 0.f, a10 = 0.f, a11 = 0.f;
        #pragma unroll 4
        for (int kk = 0; kk < SS; kk += 2) {
            const float r0 = rcol[(size_t)kk * DD];
            const float r1 = rcol[(size_t)(kk + 1) * DD];
            a00 += sc[0][qq][kk]     * r0;
            a01 += sc[0][qq][kk + 1] * r1;
            a10 += sc[1][qq][kk]     * r0;
            a11 += sc[1][qq][kk + 1] * r1;
        }
        const size_t obase = ((size_t)h0 * SS + (q0 + qq)) * DD + dd;
        out_ans[obase]                   = ab[0][qq][dd] + a00 + a01;
        out_ans[obase + (size_t)SS * DD] = ab[1][qq][dd] + a10 + a11;
    }
}

extern "C" void kernel_launch(void* const* d_in, const int* in_sizes, int n_in,
                              void* d_out, int out_size, void* d_ws, size_t ws_size,
                              hipStream_t stream) {
    (void)in_sizes; (void)n_in; (void)d_ws; (void)ws_size; (void)out_size;
    const float* q = (const float*)d_in[0];
    const float* k = (const float*)d_in[1];
    const float* v = (const float*)d_in[2];
    const float* r = (const float*)d_in[3];
    const int*   m = (const int*)d_in[4];
    float* out_ans = (float*)d_out;                       // [1,8,512,64]
    float* out_p   = out_ans + (size_t)HH * SS * DD;      // [1,8,512,512]
    dim3 grid(SS / 16, HH / 2);
    attn_relem_kernel<<<grid, 256, 0, stream>>>(q, k, v, r, m, out_ans, out_p);
}